// DSTAGNN_block_7799660609959
// MI455X (gfx1250) — compile-verified
//
#include <hip/hip_runtime.h>
#include <math.h>

// ---------------------------------------------------------------------------
// DSTAGNN block for MI455X (gfx1250, wave32, WMMA).
// fp32 end-to-end using V_WMMA_F32_16X16X4_F32 for all large GEMMs.
// ---------------------------------------------------------------------------

#define Bc 16
#define Nn 1024
#define Tt 12
#define DM 512
#define DK 32
#define Hh 3
#define Kk 3
#define Cc 32

typedef __attribute__((ext_vector_type(2))) float v2f;
typedef __attribute__((ext_vector_type(8))) float v8f;

__device__ __forceinline__ v8f wmma_f32_4(v2f a, v2f b, v8f c) {
  // D = A(16x4) * B(4x16) + C(16x16), fp32 exact.
  return __builtin_amdgcn_wmma_f32_16x16x4_f32(false, a, false, b, (short)0, c,
                                               false, false);
}

// ---------------------------------------------------------------------------
// K1: temporal embedding.  TEmx[b,t,n] = LN_n(x[b,n,0,t] + posT[t,n])*g+b
// ---------------------------------------------------------------------------
__global__ __launch_bounds__(256) void k_tembed(const float* __restrict__ x,
                                                const float* __restrict__ posT,
                                                const float* __restrict__ gT,
                                                const float* __restrict__ bT,
                                                float* __restrict__ TE) {
  int bt = blockIdx.x;          // b*12 + t
  int b = bt / Tt, t = bt % Tt;
  int tid = threadIdx.x;
  __shared__ float red[512];
  float v[4], s = 0.f, s2 = 0.f;
  for (int i = 0; i < 4; ++i) {
    int n = tid + i * 256;
    float a = x[(b * Nn + n) * Tt + t] + posT[t * Nn + n];
    v[i] = a; s += a; s2 += a * a;
  }
  red[tid] = s; red[256 + tid] = s2;
  __syncthreads();
  for (int off = 128; off > 0; off >>= 1) {
    if (tid < off) { red[tid] += red[tid + off]; red[256 + tid] += red[256 + tid + off]; }
    __syncthreads();
  }
  float mu = red[0] * (1.f / Nn);
  float var = red[256] * (1.f / Nn) - mu * mu;
  float rs = rsqrtf(var + 1e-5f);
  for (int i = 0; i < 4; ++i) {
    int n = tid + i * 256;
    TE[bt * Nn + n] = (v[i] - mu) * rs * gT[n] + bT[n];
  }
}

// ---------------------------------------------------------------------------
// K2: temporal QKV projection via fp32 WMMA. (192x1024)@(1024x96)
// grid (12 Mtiles, 6 Ntiles, 3 weights), 1 wave per 16x16 tile.
// ---------------------------------------------------------------------------
__global__ __launch_bounds__(32) void k_tqkv(const float* __restrict__ TE,
                                             const float* __restrict__ WQ,
                                             const float* __restrict__ WK,
                                             const float* __restrict__ WV,
                                             float* __restrict__ Qt,
                                             float* __restrict__ Kt,
                                             float* __restrict__ Vt) {
  int mt = blockIdx.x, nt = blockIdx.y, z = blockIdx.z;
  const float* W = (z == 0) ? WQ : (z == 1 ? WK : WV);
  float* O = (z == 0) ? Qt : (z == 1 ? Kt : Vt);
  int l = threadIdx.x;
  int lr = l & 15, koff = (l >> 4) * 2;
  int arow = mt * 16 + lr;           // A: M = lane%16
  int col = nt * 16 + lr;            // B/D: N = lane%16
  v8f c = {0.f, 0.f, 0.f, 0.f, 0.f, 0.f, 0.f, 0.f};
  for (int k0 = 0; k0 < Nn; k0 += 4) {
    float2 av = *(const float2*)(TE + arow * Nn + k0 + koff);
    v2f a; a[0] = av.x; a[1] = av.y;
    v2f bv; bv[0] = W[(k0 + koff) * 96 + col]; bv[1] = W[(k0 + koff + 1) * 96 + col];
    c = wmma_f32_4(a, bv, c);
  }
  int rb = mt * 16 + (l >> 4) * 8;   // D: rows r / 8+r
  for (int r = 0; r < 8; ++r) O[(rb + r) * 96 + col] = c[r];
}

// ---------------------------------------------------------------------------
// K3: 12x12 temporal attention per (b,h); writes pre-softmax scores (re_At)
// to d_out tail and context into CTX.
// ---------------------------------------------------------------------------
__global__ __launch_bounds__(160) void k_tattn(const float* __restrict__ Qt,
                                               const float* __restrict__ Kt,
                                               const float* __restrict__ Vt,
                                               const float* __restrict__ resA,
                                               float* __restrict__ CTX,
                                               float* __restrict__ reAt) {
  int bh = blockIdx.x;
  int b = bh / Hh, h = bh % Hh;
  int tid = threadIdx.x;
  __shared__ float S[12][12];
  __shared__ float P[12][12];
  if (tid < 144) {
    int q = tid / 12, kq = tid % 12;
    const float* qr = Qt + (b * Tt + q) * 96 + h * DK;
    const float* kr = Kt + (b * Tt + kq) * 96 + h * DK;
    float s = 0.f;
    for (int d = 0; d < DK; ++d) s = fmaf(qr[d], kr[d], s);
    s = s * 0.17677669529663687f + resA[(b * Hh + h) * 144 + q * 12 + kq];
    S[q][kq] = s;
    reAt[(b * Hh + h) * 144 + q * 12 + kq] = s;
  }
  __syncthreads();
  if (tid < 12) {
    int q = tid;
    float m = -3.0e38f;
    for (int kq = 0; kq < 12; ++kq) m = fmaxf(m, S[q][kq]);
    float sum = 0.f;
    for (int kq = 0; kq < 12; ++kq) { float e = __expf(S[q][kq] - m); P[q][kq] = e; sum += e; }
    float inv = 1.f / sum;
    for (int kq = 0; kq < 12; ++kq) P[q][kq] *= inv;
  }
  __syncthreads();
  for (int i = tid; i < 12 * DK; i += 160) {
    int q = i / DK, d = i % DK;
    float s = 0.f;
    for (int kq = 0; kq < 12; ++kq)
      s = fmaf(P[q][kq], Vt[(b * Tt + kq) * 96 + h * DK + d], s);
    CTX[(b * Tt + q) * 96 + h * DK + d] = s;
  }
}

// ---------------------------------------------------------------------------
// K4: TATout[bt,n] = LN_n(CTX[bt,:]@fc_t[:,n] + TEmx[bt,n])  (gamma=1,beta=0)
// ---------------------------------------------------------------------------
__global__ __launch_bounds__(256) void k_tout(const float* __restrict__ CTX,
                                              const float* __restrict__ fct,
                                              const float* __restrict__ TE,
                                              float* __restrict__ TAT) {
  int bt = blockIdx.x;
  int tid = threadIdx.x;
  __shared__ float cx[96];
  __shared__ float red[512];
  if (tid < 96) cx[tid] = CTX[bt * 96 + tid];
  __syncthreads();
  float v[4], s = 0.f, s2 = 0.f;
  for (int i = 0; i < 4; ++i) {
    int n = tid + i * 256;
    float a = TE[bt * Nn + n];
    for (int j = 0; j < 96; ++j) a = fmaf(cx[j], fct[j * Nn + n], a);
    v[i] = a; s += a; s2 += a * a;
  }
  red[tid] = s; red[256 + tid] = s2;
  __syncthreads();
  for (int off = 128; off > 0; off >>= 1) {
    if (tid < off) { red[tid] += red[tid + off]; red[256 + tid] += red[256 + tid + off]; }
    __syncthreads();
  }
  float mu = red[0] * (1.f / Nn);
  float var = red[256] * (1.f / Nn) - mu * mu;
  float rs = rsqrtf(var + 1e-5f);
  for (int i = 0; i < 4; ++i) {
    int n = tid + i * 256;
    TAT[bt * Nn + n] = (v[i] - mu) * rs;
  }
}

// ---------------------------------------------------------------------------
// K5: pre_conv over T (+bias, +pos_embed_S), LayerNorm over d=512 -> SEmx
// ---------------------------------------------------------------------------
__global__ __launch_bounds__(256) void k_sembed(const float* __restrict__ TAT,
                                                const float* __restrict__ pcw,
                                                const float* __restrict__ pcb,
                                                const float* __restrict__ posS,
                                                const float* __restrict__ gS,
                                                const float* __restrict__ bS,
                                                float* __restrict__ SE) {
  int bn = blockIdx.x;
  int b = bn >> 10, n = bn & 1023;
  int tid = threadIdx.x;
  __shared__ float tl[12];
  __shared__ float red[512];
  if (tid < 12) tl[tid] = TAT[(b * Tt + tid) * Nn + n];
  __syncthreads();
  float v[2], s = 0.f, s2 = 0.f;
  for (int i = 0; i < 2; ++i) {
    int d = tid + i * 256;
    float a = pcb[d] + posS[n * DM + d];
    for (int t = 0; t < Tt; ++t) a = fmaf(tl[t], pcw[d * Tt + t], a);
    v[i] = a; s += a; s2 += a * a;
  }
  red[tid] = s; red[256 + tid] = s2;
  __syncthreads();
  for (int off = 128; off > 0; off >>= 1) {
    if (tid < off) { red[tid] += red[tid + off]; red[256 + tid] += red[256 + tid + off]; }
    __syncthreads();
  }
  float mu = red[0] * (1.f / DM);
  float var = red[256] * (1.f / DM) - mu * mu;
  float rs = rsqrtf(var + 1e-5f);
  for (int i = 0; i < 2; ++i) {
    int d = tid + i * 256;
    SE[((long)bn) * DM + d] = (v[i] - mu) * rs * gS[d] + bS[d];
  }
}

// ---------------------------------------------------------------------------
// K6: spatial Q/K projection via fp32 WMMA. (16384x512)@(512x96).
// Output scattered to (B, Khead, N, 32); Q pre-scaled by 1/sqrt(DK).
// ---------------------------------------------------------------------------
__global__ __launch_bounds__(32) void k_sqk(const float* __restrict__ SE,
                                            const float* __restrict__ WQs,
                                            const float* __restrict__ WKs,
                                            float* __restrict__ Qs,
                                            float* __restrict__ Ks) {
  int mt = blockIdx.x, nt = blockIdx.y, z = blockIdx.z;
  const float* W = z ? WKs : WQs;
  float* O = z ? Ks : Qs;
  float scale = z ? 1.0f : 0.17677669529663687f;  // fold 1/sqrt(32) into Q
  int l = threadIdx.x;
  int lr = l & 15, koff = (l >> 4) * 2;
  int arow = mt * 16 + lr;
  int col = nt * 16 + lr;
  v8f c = {0.f, 0.f, 0.f, 0.f, 0.f, 0.f, 0.f, 0.f};
  for (int k0 = 0; k0 < DM; k0 += 4) {
    float2 av = *(const float2*)(SE + (long)arow * DM + k0 + koff);
    v2f a; a[0] = av.x; a[1] = av.y;
    v2f bv; bv[0] = W[(k0 + koff) * 96 + col]; bv[1] = W[(k0 + koff + 1) * 96 + col];
    c = wmma_f32_4(a, bv, c);
  }
  int kh = col >> 5, d = col & 31;
  int rb = mt * 16 + (l >> 4) * 8;
  for (int r = 0; r < 8; ++r) {
    int rg = rb + r;
    int b = rg >> 10, n = rg & 1023;
    O[(((long)(b * Kk + kh) << 10) + n) * 32 + d] = c[r] * scale;
  }
}

// ---------------------------------------------------------------------------
// K7: fused spatial attention + Chebyshev aggregation (flash style).
// One wave per (b, khead, 16 output columns n).  Online softmax over m
// (the contraction index), score tiles 16x16 via fp32 WMMA.
// rhs[b,k,n,t] = sum_m softmax_m(Q_m.K_n + adj*mask) * cheb[k,m,n] * x[b,m,t]
// ---------------------------------------------------------------------------
__global__ __launch_bounds__(32) void k_spatial_fused(
    const float* __restrict__ Qs, const float* __restrict__ Ks,
    const float* __restrict__ x, const float* __restrict__ cheb,
    const float* __restrict__ adj, const float* __restrict__ cmask,
    float* __restrict__ rhs) {
  __shared__ float xs[16 * Tt];
  int l = threadIdx.x;
  int n0 = blockIdx.x * 16;
  int kh = blockIdx.y;
  int b = blockIdx.z;
  int lr = l & 15, hi = l >> 4, koff = hi * 2;
  int n = n0 + lr;
  long bk = (long)(b * Kk + kh);

  // Preload B fragments (Ks row for this lane's column), all 8 k-steps.
  const float* Krow = Ks + (bk * Nn + n) * 32;
  float bb0[8], bb1[8];
  for (int kk = 0; kk < 8; ++kk) {
    float2 t2 = *(const float2*)(Krow + kk * 4 + koff);
    bb0[kk] = t2.x; bb1[kk] = t2.y;
  }

  float runM = -3.0e38f, runL = 0.f;
  float acc[Tt];
  for (int t = 0; t < Tt; ++t) acc[t] = 0.f;

  const float* Qbase = Qs + bk * Nn * 32;
  const float* chebK = cheb + (long)kh * Nn * Nn;
  const float* maskK = cmask + (long)kh * Nn * Nn;
  const float* xb = x + (long)b * Nn * Tt;

  for (int mi = 0; mi < Nn / 16; ++mi) {
    int mb = mi * 16;
    __syncthreads();
    for (int i = 0; i < 6; ++i) {        // stage x[b, mb..mb+15, 0..11]
      int idx = l * 6 + i;
      xs[idx] = xb[mb * Tt + idx];
    }
    __syncthreads();

    // Score tile: S(16m x 16n) = Qtile(16x32) * KsTile^T, K in 8 wmma steps.
    v8f cc = {0.f, 0.f, 0.f, 0.f, 0.f, 0.f, 0.f, 0.f};
    const float* Qrow = Qbase + (long)(mb + lr) * 32;
    for (int kk = 0; kk < 8; ++kk) {
      float2 av = *(const float2*)(Qrow + kk * 4 + koff);
      v2f a; a[0] = av.x; a[1] = av.y;
      v2f bv; bv[0] = bb0[kk]; bv[1] = bb1[kk];
      cc = wmma_f32_4(a, bv, cc);
    }

    // Lane owns column n, rows m = mb + hi*8 + r.
    int rbase = mb + hi * 8;
    float sv[8], mloc = -3.0e38f;
    for (int r = 0; r < 8; ++r) {
      int m = rbase + r;
      float bias = adj[(long)m * Nn + n] * maskK[(long)m * Nn + n];
      float s = cc[r] + bias;            // Q already carries 1/sqrt(DK)
      sv[r] = s;
      mloc = fmaxf(mloc, s);
    }
    float pm = fmaxf(mloc, __shfl_xor(mloc, 16, 32));
    float newM = fmaxf(runM, pm);
    float scl = __expf(runM - newM);
    runL *= scl;
    for (int t = 0; t < Tt; ++t) acc[t] *= scl;
    float ls = 0.f;
    for (int r = 0; r < 8; ++r) {
      int m = rbase + r;
      float p = __expf(sv[r] - newM);
      ls += p;
      float w = p * chebK[(long)m * Nn + n];
      const float* xr = &xs[(hi * 8 + r) * Tt];
      for (int t = 0; t < Tt; ++t) acc[t] = fmaf(w, xr[t], acc[t]);
    }
    runL += ls + __shfl_xor(ls, 16, 32);
    runM = newM;
  }

  float inv = 1.f / runL;
  float ov[Tt];
  for (int t = 0; t < Tt; ++t)
    ov[t] = (acc[t] + __shfl_xor(acc[t], 16, 32)) * inv;
  if (l < 16) {
    float* orow = rhs + (bk * Nn + n) * Tt;
    for (int t = 0; t < Tt; ++t) orow[t] = ov[t];
  }
}

// ---------------------------------------------------------------------------
// K8: Theta combine + GTU gated convs + fcmy + residual + channel LayerNorm.
// One block per (b,n).
// ---------------------------------------------------------------------------
__global__ __launch_bounds__(128) void k_final(
    const float* __restrict__ rhs, const float* __restrict__ Theta,
    const float* __restrict__ x,
    const float* __restrict__ g3w, const float* __restrict__ g3b,
    const float* __restrict__ g5w, const float* __restrict__ g5b,
    const float* __restrict__ g7w, const float* __restrict__ g7b,
    const float* __restrict__ rcw, const float* __restrict__ rcb,
    const float* __restrict__ fw, const float* __restrict__ fb,
    const float* __restrict__ gf, const float* __restrict__ bf,
    float* __restrict__ out) {
  int bn = blockIdx.x;
  int b = bn >> 10, n = bn & 1023;
  int tid = threadIdx.x;
  __shared__ float X[Cc][Tt];
  __shared__ float cat[Cc][24];
  __shared__ float V[Cc][Tt];
  __shared__ float mu[Tt], rsd[Tt];

  // A: spatial_gcn = relu(sum_k rhs*Theta)
  for (int i = tid; i < Cc * Tt; i += 128) {
    int c = i / Tt, t = i % Tt;
    float s = 0.f;
    for (int k = 0; k < Kk; ++k)
      s = fmaf(rhs[(((long)(b * Kk + k) << 10) + n) * Tt + t], Theta[k * Cc + c], s);
    X[c][t] = fmaxf(s, 0.f);
  }
  __syncthreads();

  // B: three GTU branches, gated and concatenated along time (10+8+6=24)
  for (int i = tid; i < Cc * 24; i += 128) {
    int c = i / 24, tc = i % 24;
    int ks, pos; const float* w; const float* bs;
    if (tc < 10)      { ks = 3; pos = tc;      w = g3w; bs = g3b; }
    else if (tc < 18) { ks = 5; pos = tc - 10; w = g5w; bs = g5b; }
    else              { ks = 7; pos = tc - 18; w = g7w; bs = g7b; }
    float y1 = bs[c], y2 = bs[c + Cc];
    const float* w1 = w + (long)c * Cc * ks;
    const float* w2 = w + (long)(c + Cc) * Cc * ks;
    for (int ci = 0; ci < Cc; ++ci)
      for (int j = 0; j < ks; ++j) {
        float xv = X[ci][pos + j];
        y1 = fmaf(xv, w1[ci * ks + j], y1);
        y2 = fmaf(xv, w2[ci * ks + j], y2);
      }
    cat[c][tc] = tanhf(y1) * (1.f / (1.f + __expf(-y2)));
  }
  __syncthreads();

  // C: fcmy projection (24->12), relu, residual 1x1 conv, relu
  const float* xr = x + ((long)b * Nn + n) * Tt;
  for (int i = tid; i < Cc * Tt; i += 128) {
    int c = i / Tt, t = i % Tt;
    float s = fb[t];
    for (int tc = 0; tc < 24; ++tc) s = fmaf(cat[c][tc], fw[tc * Tt + t], s);
    float tco = fmaxf(s, 0.f);
    float res = fmaf(xr[t], rcw[c], rcb[c]);
    V[c][t] = fmaxf(res + tco, 0.f);
  }
  __syncthreads();

  // D: LayerNorm over channel dim (32) per t
  if (tid < Tt) {
    float s = 0.f, s2 = 0.f;
    for (int c = 0; c < Cc; ++c) { float v = V[c][tid]; s += v; s2 += v * v; }
    float m = s * (1.f / Cc);
    float var = s2 * (1.f / Cc) - m * m;
    mu[tid] = m; rsd[tid] = rsqrtf(var + 1e-5f);
  }
  __syncthreads();
  for (int i = tid; i < Cc * Tt; i += 128) {
    int c = i / Tt, t = i % Tt;
    out[(((long)bn) * Cc + c) * Tt + t] = (V[c][t] - mu[t]) * rsd[t] * gf[c] + bf[c];
  }
}

// ---------------------------------------------------------------------------
extern "C" void kernel_launch(void* const* d_in, const int* in_sizes, int n_in,
                              void* d_out, int out_size, void* d_ws,
                              size_t ws_size, hipStream_t stream) {
  (void)in_sizes; (void)n_in; (void)out_size; (void)ws_size;
  const float* x    = (const float*)d_in[0];
  const float* resA = (const float*)d_in[1];
  const float* posT = (const float*)d_in[2];
  const float* gT   = (const float*)d_in[3];
  const float* bT   = (const float*)d_in[4];
  const float* WQt  = (const float*)d_in[5];
  const float* WKt  = (const float*)d_in[6];
  const float* WVt  = (const float*)d_in[7];
  const float* fct  = (const float*)d_in[8];
  const float* pcw  = (const float*)d_in[9];
  const float* pcb  = (const float*)d_in[10];
  const float* posS = (const float*)d_in[11];
  const float* gS   = (const float*)d_in[12];
  const float* bS   = (const float*)d_in[13];
  const float* WQs  = (const float*)d_in[14];
  const float* WKs  = (const float*)d_in[15];
  const float* cheb = (const float*)d_in[16];
  const float* adj  = (const float*)d_in[17];
  const float* cmsk = (const float*)d_in[18];
  const float* Th   = (const float*)d_in[19];
  const float* g3w  = (const float*)d_in[20];
  const float* g3b  = (const float*)d_in[21];
  const float* g5w  = (const float*)d_in[22];
  const float* g5b  = (const float*)d_in[23];
  const float* g7w  = (const float*)d_in[24];
  const float* g7b  = (const float*)d_in[25];
  const float* rcw  = (const float*)d_in[26];
  const float* rcb  = (const float*)d_in[27];
  const float* fw   = (const float*)d_in[28];
  const float* fb   = (const float*)d_in[29];
  const float* gf   = (const float*)d_in[30];
  const float* bf   = (const float*)d_in[31];
  float* out = (float*)d_out;
  float* ws  = (float*)d_ws;

  // workspace layout (floats)
  float* TE  = ws;                         // 192*1024
  float* Qt  = ws + 196608;                // 192*96
  float* Kt  = Qt + 18432;
  float* Vt  = Kt + 18432;
  float* CTX = Vt + 18432;                 // 192*96
  float* TAT = CTX + 18432;                // 192*1024
  float* SE  = TAT + 196608;               // 16384*512
  float* Qsp = SE + 8388608;               // 16*3*1024*32
  float* Ksp = Qsp + 1572864;
  float* RHS = Ksp + 1572864;              // 16*3*1024*12

  float* reAt = out + (long)Bc * Nn * Cc * Tt;  // tail of d_out

  k_tembed<<<Bc * Tt, 256, 0, stream>>>(x, posT, gT, bT, TE);
  k_tqkv<<<dim3(12, 6, 3), 32, 0, stream>>>(TE, WQt, WKt, WVt, Qt, Kt, Vt);
  k_tattn<<<Bc * Hh, 160, 0, stream>>>(Qt, Kt, Vt, resA, CTX, reAt);
  k_tout<<<Bc * Tt, 256, 0, stream>>>(CTX, fct, TE, TAT);
  k_sembed<<<Bc * Nn, 256, 0, stream>>>(TAT, pcw, pcb, posS, gS, bS, SE);
  k_sqk<<<dim3(Bc * Nn / 16, 6, 2), 32, 0, stream>>>(SE, WQs, WKs, Qsp, Ksp);
  k_spatial_fused<<<dim3(Nn / 16, Kk, Bc), 32, 0, stream>>>(Qsp, Ksp, x, cheb,
                                                            adj, cmsk, RHS);
  k_final<<<Bc * Nn, 128, 0, stream>>>(RHS, Th, x, g3w, g3b, g5w, g5b, g7w,
                                       g7b, rcw, rcb, fw, fb, gf, bf, out);
}